// DGCNN_Model_5643587027209
// MI455X (gfx1250) — compile-verified
//
#include <hip/hip_runtime.h>
#include <cstddef>

typedef float v2f __attribute__((ext_vector_type(2)));
typedef float v8f __attribute__((ext_vector_type(8)));

#define NN      62
#define FF      5
#define HH      64
#define KREAL   310      // NN*FF
#define KPAD    312      // padded to multiple of 4
#define BTOT    32768
#define ROWS_PB 128
#define NPART   512

// ---------------- K1: per-feature sum / sumsq partials (deterministic) ----------
__global__ __launch_bounds__(256) void bn_partial(const float* __restrict__ X,
                                                  float* __restrict__ part, int R) {
  float s[5] = {0, 0, 0, 0, 0};
  float q[5] = {0, 0, 0, 0, 0};
  int stride = gridDim.x * blockDim.x;
  for (int r = blockIdx.x * blockDim.x + threadIdx.x; r < R; r += stride) {
    const float* p = X + (size_t)r * 5;
#pragma unroll
    for (int f = 0; f < 5; ++f) {
      float v = p[f];
      s[f] += v;
      q[f] += v * v;
    }
  }
#pragma unroll
  for (int f = 0; f < 5; ++f) {
    for (int off = 16; off > 0; off >>= 1) {
      s[f] += __shfl_down(s[f], off, 32);
      q[f] += __shfl_down(q[f], off, 32);
    }
  }
  __shared__ float red[8][10];
  int lane = threadIdx.x & 31, wid = threadIdx.x >> 5;
  if (lane == 0) {
#pragma unroll
    for (int f = 0; f < 5; ++f) {
      red[wid][f] = s[f];
      red[wid][5 + f] = q[f];
    }
  }
  __syncthreads();
  if (threadIdx.x < 10) {
    float t = 0.f;
#pragma unroll
    for (int w = 0; w < 8; ++w) t += red[w][threadIdx.x];
    part[blockIdx.x * 10 + threadIdx.x] = t;
  }
}

// ---------------- K2: finalize BN scale/shift -----------------------------------
__global__ void bn_final(const float* __restrict__ part, const float* __restrict__ gamma,
                         const float* __restrict__ beta, float* __restrict__ stats,
                         int P, float invR) {
  int f = threadIdx.x;
  if (f < 5) {
    float s = 0.f, q = 0.f;
    for (int p = 0; p < P; ++p) {
      s += part[p * 10 + f];
      q += part[p * 10 + 5 + f];
    }
    float mean = s * invR;
    float var = q * invR - mean * mean;
    float scale = gamma[f] / sqrtf(var + 1e-5f);
    stats[f] = scale;               // BN scale per feature
    stats[5 + f] = beta[f] - mean * scale;  // BN shift per feature
  }
}

// ---------------- K3: build normalized S then S^2 -------------------------------
__global__ __launch_bounds__(256) void build_S2(const float* __restrict__ tril,
                                                float* __restrict__ S2out) {
  __shared__ float M[NN * 63];
  __shared__ float dv[NN];
  int tid = threadIdx.x;
  // symmetric W from lower-tri params, then ReLU
  for (int idx = tid; idx < NN * NN; idx += 256) {
    int i = idx / NN, j = idx - (idx / NN) * NN;
    int I = i > j ? i : j, J = i > j ? j : i;
    float w = tril[I * (I + 1) / 2 + J];
    M[i * 63 + j] = fmaxf(w, 0.f);
  }
  __syncthreads();
  if (tid < NN) {
    float d = 0.f;
    for (int j = 0; j < NN; ++j) d += M[tid * 63 + j];
    dv[tid] = 1.0f / sqrtf(d + 1e-10f);
  }
  __syncthreads();
  for (int idx = tid; idx < NN * NN; idx += 256) {
    int i = idx / NN, j = idx - (idx / NN) * NN;
    M[i * 63 + j] *= dv[i] * dv[j];   // L
  }
  __syncthreads();
  if (tid < NN) {
    float d = 0.f;
    for (int j = 0; j < NN; ++j) d += fabsf(M[tid * 63 + j]);
    dv[tid] = 1.0f / sqrtf(d + 1.0f);  // +1 self loop
  }
  __syncthreads();
  for (int idx = tid; idx < NN * NN; idx += 256) {
    int i = idx / NN, j = idx - (idx / NN) * NN;
    float l = M[i * 63 + j] + (i == j ? 1.0f : 0.0f);
    M[i * 63 + j] = dv[i] * l * dv[j];  // S
  }
  __syncthreads();
  for (int idx = tid; idx < NN * NN; idx += 256) {
    int i = idx / NN, j = idx - (idx / NN) * NN;
    float acc = 0.f;
    for (int k = 0; k < NN; ++k) acc += M[i * 63 + k] * M[k * 63 + j];
    S2out[i * NN + j] = acc;            // S2[n][m]
  }
}

// ---------------- K4: W2[n,f,o] = sum_h linW[f,h]*fc1W[n*64+h,o] ----------------
__global__ void calc_W2(const float* __restrict__ linW, const float* __restrict__ fc1W,
                        float* __restrict__ W2) {
  int n = blockIdx.x;
  int f = threadIdx.x / 64;
  int o = threadIdx.x & 63;
  float acc = 0.f;
  for (int h = 0; h < 64; ++h) acc += linW[f * 64 + h] * fc1W[(size_t)(n * 64 + h) * 64 + o];
  W2[(size_t)(n * 5 + f) * 64 + o] = acc;
}

// --------- K5: W3[m,f,o] = sum_n S2[n,m]*W2[n,f,o]; WT[o][k]=scale_f*W3 ---------
__global__ void calc_W3(const float* __restrict__ S2, const float* __restrict__ W2,
                        const float* __restrict__ stats, float* __restrict__ W3,
                        float* __restrict__ WT) {
  int id = blockIdx.x * blockDim.x + threadIdx.x;   // 62*64 threads
  int m = id >> 6, o = id & 63;
  if (m >= NN) return;
#pragma unroll
  for (int f = 0; f < 5; ++f) {
    float acc = 0.f;
    for (int n = 0; n < NN; ++n) acc += S2[n * NN + m] * W2[(size_t)(n * 5 + f) * 64 + o];
    W3[(size_t)(m * 5 + f) * 64 + o] = acc;
    WT[(size_t)o * KPAD + m * 5 + f] = stats[f] * acc;
  }
}

// ---------------- K6: fused bias2[o] + zero K-pad of WT -------------------------
__global__ void calc_bias(const float* __restrict__ fc1b, const float* __restrict__ linb,
                          const float* __restrict__ fc1W, const float* __restrict__ stats,
                          const float* __restrict__ W3, float* __restrict__ bias2,
                          float* __restrict__ WT) {
  int o = threadIdx.x;  // 64 threads
  float b = fc1b[o];
  for (int n = 0; n < NN; ++n)
    for (int h = 0; h < 64; ++h) b += linb[h] * fc1W[(size_t)(n * 64 + h) * 64 + o];
  for (int kk = 0; kk < KREAL; ++kk) b += stats[5 + kk % 5] * W3[(size_t)kk * 64 + o];
  bias2[o] = b;
  WT[(size_t)o * KPAD + 310] = 0.f;
  WT[(size_t)o * KPAD + 311] = 0.f;
}

// ---------------- K7: fused (B x 310)x(310 x 64) WMMA GEMM + ReLU + fc2 ---------
__global__ __launch_bounds__(256) void sgc_fused(const float* __restrict__ X,
                                                 const float* __restrict__ WT,
                                                 const float* __restrict__ bias2,
                                                 const float* __restrict__ fc2W,
                                                 const float* __restrict__ fc2b,
                                                 float* __restrict__ out) {
  extern __shared__ float sX[];  // ROWS_PB * KPAD floats, zero-padded K tail
  const int tid = threadIdx.x;
  const int gRow0 = blockIdx.x * ROWS_PB;

  // Cooperative coalesced load of padded X tile
  for (int idx = tid; idx < ROWS_PB * KPAD; idx += 256) {
    int r = idx / KPAD;
    int k = idx - r * KPAD;
    sX[idx] = (k < KREAL) ? X[(size_t)(gRow0 + r) * KREAL + k] : 0.f;
  }
  __syncthreads();

  const int lane = tid & 31, wid = tid >> 5;
  const int lh = lane & 15;       // M (or N) index within 16
  const int kh = lane >> 4;       // K half: 0 -> K=k,k+1 ; 1 -> K=k+2,k+3

  const v2f* ap = (const v2f*)(sX + (wid * 16 + lh) * KPAD + 2 * kh);
  const v2f* bp = (const v2f*)(WT + (size_t)lh * KPAD + 2 * kh);

  v8f c0 = {0, 0, 0, 0, 0, 0, 0, 0};
  v8f c1 = {0, 0, 0, 0, 0, 0, 0, 0};
  v8f c2 = {0, 0, 0, 0, 0, 0, 0, 0};
  v8f c3 = {0, 0, 0, 0, 0, 0, 0, 0};

#pragma unroll 2
  for (int i = 0; i < KPAD / 4; ++i) {     // 78 k-steps of 4
    v2f a = ap[2 * i];
    v2f b0 = bp[2 * i];
    v2f b1 = bp[2 * i + 16 * KPAD / 2];    // N tile 1 (+19968 B imm offset)
    v2f b2 = bp[2 * i + 32 * KPAD / 2];    // N tile 2
    v2f b3 = bp[2 * i + 48 * KPAD / 2];    // N tile 3
    c0 = __builtin_amdgcn_wmma_f32_16x16x4_f32(false, a, false, b0, (short)0, c0, false, false);
    c1 = __builtin_amdgcn_wmma_f32_16x16x4_f32(false, a, false, b1, (short)0, c1, false, false);
    c2 = __builtin_amdgcn_wmma_f32_16x16x4_f32(false, a, false, b2, (short)0, c2, false, false);
    c3 = __builtin_amdgcn_wmma_f32_16x16x4_f32(false, a, false, b3, (short)0, c3, false, false);
  }

  // Spill z tile (16 rows x 64) into this wave's own LDS region (safe reuse)
  float* zb = sX + wid * 16 * KPAD;
#pragma unroll
  for (int v = 0; v < 8; ++v) {
    int row = v + 8 * kh;
    zb[row * 64 + 0  + lh] = c0[v];
    zb[row * 64 + 16 + lh] = c1[v];
    zb[row * 64 + 32 + lh] = c2[v];
    zb[row * 64 + 48 + lh] = c3[v];
  }

  // bias + ReLU + 64x3 fc2; 16 lanes each own one batch row
  if (lane < 16) {
    int row = lh;
    float a0 = fc2b[0], a1 = fc2b[1], a2 = fc2b[2];
    for (int o = 0; o < 64; ++o) {
      float z = fmaxf(zb[row * 64 + o] + bias2[o], 0.f);
      a0 += z * fc2W[o * 3 + 0];
      a1 += z * fc2W[o * 3 + 1];
      a2 += z * fc2W[o * 3 + 2];
    }
    size_t ob = (size_t)(gRow0 + wid * 16 + row) * 3;
    out[ob + 0] = a0;
    out[ob + 1] = a1;
    out[ob + 2] = a2;
  }
}

extern "C" void kernel_launch(void* const* d_in, const int* in_sizes, int n_in,
                              void* d_out, int out_size, void* d_ws, size_t ws_size,
                              hipStream_t stream) {
  const float* X     = (const float*)d_in[0];
  const float* tril  = (const float*)d_in[1];
  const float* gamma = (const float*)d_in[2];
  const float* beta  = (const float*)d_in[3];
  const float* linW  = (const float*)d_in[4];
  const float* linb  = (const float*)d_in[5];
  const float* fc1W  = (const float*)d_in[6];
  const float* fc1b  = (const float*)d_in[7];
  const float* fc2W  = (const float*)d_in[8];
  const float* fc2b  = (const float*)d_in[9];
  float* out = (float*)d_out;
  float* w = (float*)d_ws;

  float* part  = w;          // NPART*10 = 5120
  float* stats = w + 5120;   // 10 (scale[5], shift[5])
  float* S2    = w + 5136;   // 3844
  float* W2    = w + 8980;   // 19840
  float* W3    = w + 28820;  // 19840
  float* WT    = w + 48660;  // 64*312 = 19968
  float* bias2 = w + 68628;  // 64

  const int R = BTOT * NN;

  bn_partial<<<NPART, 256, 0, stream>>>(X, part, R);
  bn_final<<<1, 32, 0, stream>>>(part, gamma, beta, stats, NPART, 1.0f / (float)R);
  build_S2<<<1, 256, 0, stream>>>(tril, S2);
  calc_W2<<<NN, 320, 0, stream>>>(linW, fc1W, W2);
  calc_W3<<<31, 128, 0, stream>>>(S2, W2, stats, W3, WT);  // 31*128 = 62*64
  calc_bias<<<1, 64, 0, stream>>>(fc1b, linb, fc1W, stats, W3, bias2, WT);

  const int lds_bytes = ROWS_PB * KPAD * (int)sizeof(float);  // 159744 B < 320 KB WGP LDS
  (void)hipFuncSetAttribute((const void*)sgc_fused,
                            hipFuncAttributeMaxDynamicSharedMemorySize, lds_bytes);
  sgc_fused<<<BTOT / ROWS_PB, 256, lds_bytes, stream>>>(X, WT, bias2, fc2W, fc2b, out);
}